// CMBPS_74646531604674
// MI455X (gfx1250) — compile-verified
//
#include <hip/hip_runtime.h>

typedef float v2f __attribute__((ext_vector_type(2)));
typedef float v8f __attribute__((ext_vector_type(8)));

#define H_KNOT    1.41421356237309504880f   // sqrt(2): uniform knot spacing
#define INV_H     0.70710678118654752440f   // 1/sqrt(2)
#define NORM_CONST (1.0f / 12661.0f)
#define BATCH 512
#define NP    112    // hidden dim 100 padded to multiple of 16
#define N4    128    // final layer width

// ---------------------------------------------------------------------------
// Kernel 1: fused 4-layer MLP via V_WMMA_F32_16X16X4_F32.
// grid = 8 blocks x 128 threads (4 wave32 per block, 16 batch rows per wave).
// Weights staged in LDS (padded, zero-filled); activations double-buffered in
// LDS per wave; final layer stored TRANSPOSED to y[128][512].
// ---------------------------------------------------------------------------
__global__ __launch_bounds__(128) void mlp_wmma_kernel(
    const float* __restrict__ phi,
    const float* __restrict__ W1, const float* __restrict__ b1,
    const float* __restrict__ W2, const float* __restrict__ b2,
    const float* __restrict__ W3, const float* __restrict__ b3,
    const float* __restrict__ W4, const float* __restrict__ b4,
    float* __restrict__ y /* [128][512] */)
{
  extern __shared__ float smem[];
  float* sW1 = smem;                 // 4 * 112
  float* sW2 = sW1 + 4 * NP;         // 112 * 112
  float* sW3 = sW2 + NP * NP;        // 112 * 112
  float* sW4 = sW3 + NP * NP;        // 112 * 128
  float* sB1 = sW4 + NP * N4;        // 112
  float* sB2 = sB1 + NP;             // 112
  float* sB3 = sB2 + NP;             // 112
  float* sB4 = sB3 + NP;             // 128
  float* sH  = sB4 + N4;             // 4 waves * 2 bufs * 16*112

  const int tid = threadIdx.x;

  // ---- cooperative weight staging (zero padded) ----
  for (int i = tid; i < 4 * NP; i += 128) {
    int k = i / NP, n = i % NP;
    sW1[i] = (k < 2 && n < 100) ? W1[k * 100 + n] : 0.0f;
  }
  for (int i = tid; i < NP * NP; i += 128) {
    int k = i / NP, n = i % NP;
    bool ok = (k < 100 && n < 100);
    sW2[i] = ok ? W2[k * 100 + n] : 0.0f;
    sW3[i] = ok ? W3[k * 100 + n] : 0.0f;
  }
  for (int i = tid; i < NP * N4; i += 128) {
    int k = i / N4, n = i % N4;
    sW4[i] = (k < 100) ? W4[k * N4 + n] : 0.0f;
  }
  for (int i = tid; i < NP; i += 128) {
    sB1[i] = (i < 100) ? b1[i] : 0.0f;
    sB2[i] = (i < 100) ? b2[i] : 0.0f;
    sB3[i] = (i < 100) ? b3[i] : 0.0f;
  }
  for (int i = tid; i < N4; i += 128) sB4[i] = b4[i];
  __syncthreads();

  const int wave = tid >> 5;
  const int lane = tid & 31;
  const int col  = lane & 15;  // N within tile; also A-matrix row M
  const int hi   = lane >> 4;  // upper half of wave -> K+2 / M+8 per ISA layout
  const int rowbase = blockIdx.x * 64 + wave * 16;

  float* hA = sH + (wave * 2 + 0) * (16 * NP);
  float* hB = sH + (wave * 2 + 1) * (16 * NP);

  // ---- layer 1 input: x = (phi*dphi + min - mu)/sig ----
  float x0, x1;
  {
    int r = rowbase + col;
    float p0 = phi[r * 2 + 0];
    float p1 = phi[r * 2 + 1];
    x0 = 2.0f * p0 - 1.0f;                       // (40p+50-70)/20
    x1 = (p1 * 0.0492f - 0.0245f) * 40.0f;       // (.0492p+.0075-.032)/.025
  }

  // ---- layer 1: 16x112 = (16x4 zero-padded) @ (4x112), one WMMA per n-tile
  for (int nt = 0; nt < 7; ++nt) {
    v8f acc;
    float bb = sB1[nt * 16 + col];
#pragma unroll
    for (int v = 0; v < 8; ++v) acc[v] = bb;
    v2f a, b;
    a.x = hi ? 0.0f : x0;          // K=2,3 rows are zero pad
    a.y = hi ? 0.0f : x1;
    int kb = hi * 2;
    b.x = sW1[kb * NP + nt * 16 + col];
    b.y = sW1[(kb + 1) * NP + nt * 16 + col];
    acc = __builtin_amdgcn_wmma_f32_16x16x4_f32(false, a, false, b,
                                                (short)0, acc, false, false);
#pragma unroll
    for (int v = 0; v < 8; ++v)
      hA[(v + hi * 8) * NP + nt * 16 + col] = fmaxf(acc[v], 0.0f);
  }

  // ---- layers 2 and 3: 16x112 @ 112x112, relu, LDS double buffer ----
  const float* wl[2] = { sW2, sW3 };
  const float* bl[2] = { sB2, sB3 };
  float* hin  = hA;
  float* hout = hB;
  for (int layer = 0; layer < 2; ++layer) {
    const float* w  = wl[layer];
    const float* bi = bl[layer];
    for (int nt = 0; nt < 7; ++nt) {
      v8f acc;
      float bb = bi[nt * 16 + col];
#pragma unroll
      for (int v = 0; v < 8; ++v) acc[v] = bb;
      for (int k0 = 0; k0 < NP; k0 += 4) {
        int kb = k0 + hi * 2;
        v2f a, b;
        a.x = hin[col * NP + kb];
        a.y = hin[col * NP + kb + 1];
        b.x = w[kb * NP + nt * 16 + col];
        b.y = w[(kb + 1) * NP + nt * 16 + col];
        acc = __builtin_amdgcn_wmma_f32_16x16x4_f32(false, a, false, b,
                                                    (short)0, acc, false, false);
      }
#pragma unroll
      for (int v = 0; v < 8; ++v)
        hout[(v + hi * 8) * NP + nt * 16 + col] = fmaxf(acc[v], 0.0f);
    }
    float* t = hin; hin = hout; hout = t;
  }

  // ---- layer 4: 16x112 @ 112x128, no relu, store transposed to y[n][batch]
  for (int nt = 0; nt < 8; ++nt) {
    v8f acc;
    float bb = sB4[nt * 16 + col];
#pragma unroll
    for (int v = 0; v < 8; ++v) acc[v] = bb;
    for (int k0 = 0; k0 < NP; k0 += 4) {
      int kb = k0 + hi * 2;
      v2f a, b;
      a.x = hin[col * NP + kb];
      a.y = hin[col * NP + kb + 1];
      b.x = sW4[kb * N4 + nt * 16 + col];
      b.y = sW4[(kb + 1) * N4 + nt * 16 + col];
      acc = __builtin_amdgcn_wmma_f32_16x16x4_f32(false, a, false, b,
                                                  (short)0, acc, false, false);
    }
#pragma unroll
    for (int v = 0; v < 8; ++v) {
      int n = nt * 16 + col;
      int m = rowbase + v + hi * 8;
      y[n * BATCH + m] = acc[v];
    }
  }
}

// ---------------------------------------------------------------------------
// Kernel 2: natural cubic spline, uniform knots h = sqrt(2).
// Tridiag(1,4,1) Thomas solve per channel (thread = channel).
// Emits packed coefficients coef[ch*127 + k] = {a,b,c,d}.
// ---------------------------------------------------------------------------
__global__ __launch_bounds__(128) void spline_kernel(
    const float* __restrict__ y /* [128][512] */,
    float4* __restrict__ coef /* [512][127] */)
{
  extern __shared__ float smem[];
  float* dp = smem;            // 126 * 128
  float* cp = dp + 126 * 128;  // 127

  const int tid = threadIdx.x;
  const int ch  = blockIdx.x * 128 + tid;

  if (tid == 0) {
    float c = 0.0f;
    cp[0] = 0.0f;
    for (int k = 1; k <= 126; ++k) { c = 1.0f / (4.0f - c); cp[k] = c; }
  }
  __syncthreads();

  // forward sweep: r_k = (6/h^2)(y[k+1]-2y[k]+y[k-1]) = 3*(...), h^2 = 2
  float yprev = y[0 * BATCH + ch];
  float ycur  = y[1 * BATCH + ch];
  float dprev = 0.0f;
  for (int k = 1; k <= 126; ++k) {
    float ynext = y[(k + 1) * BATCH + ch];
    float r  = 3.0f * (ynext - 2.0f * ycur + yprev);
    float dk = (r - dprev) * cp[k];
    dp[(k - 1) * 128 + tid] = dk;
    dprev = dk; yprev = ycur; ycur = ynext;
  }

  // backward sweep + coefficient emission (M_0 = M_127 = 0)
  float Mnext = 0.0f;
  float ykp1  = y[127 * BATCH + ch];
  for (int k = 126; k >= 0; --k) {
    float Mk = (k == 0) ? 0.0f : (dp[(k - 1) * 128 + tid] - cp[k] * Mnext);
    float yk = y[k * BATCH + ch];
    float4 cf;
    cf.x = yk;
    cf.y = (ykp1 - yk) * INV_H - H_KNOT * (2.0f * Mk + Mnext) * (1.0f / 6.0f);
    cf.z = 0.5f * Mk;
    cf.w = (Mnext - Mk) * (1.0f / (6.0f * H_KNOT));
    coef[ch * 127 + k] = cf;
    Mnext = Mk; ykp1 = yk;
  }
}

// ---------------------------------------------------------------------------
// Kernel 3: evaluate spline + exp over the 256x256 grid for all 512 channels.
// Uniform knots => idx = floor(w/sqrt2). One 16B coeff gather (hot in L2),
// 3 FMAs, v_exp_f32, non-temporal 4B store (128 MiB write-once output).
// ---------------------------------------------------------------------------
__global__ __launch_bounds__(256) void eval_kernel(
    const float4* __restrict__ coef, float* __restrict__ out)
{
  const int p = blockIdx.x * 256 + threadIdx.x;   // 0 .. 65535
  const int i = p >> 8, j = p & 255;
  const float fi = (float)(i < 128 ? i : i - 256);
  const float fj = (float)(j < 128 ? j : j - 256);
  const float w  = sqrtf(fi * fi + fj * fj);
  int idx = (int)(w * INV_H);
  idx = idx > 126 ? 126 : idx;
  const float frac = w - (float)idx * H_KNOT;

  const float4* cptr = coef + idx;   // + ch*127 per channel
  float* optr = out + p;
#pragma unroll 4
  for (int ch = 0; ch < 512; ++ch) {
    float4 cf = cptr[ch * 127];
    float val = fmaf(frac, fmaf(frac, fmaf(frac, cf.w, cf.z), cf.y), cf.x);
    __builtin_nontemporal_store(__expf(val) * NORM_CONST, optr + ch * 65536);
  }
}

// ---------------------------------------------------------------------------
extern "C" void kernel_launch(void* const* d_in, const int* in_sizes, int n_in,
                              void* d_out, int out_size, void* d_ws, size_t ws_size,
                              hipStream_t stream) {
  const float* phi = (const float*)d_in[0];
  const float* W1  = (const float*)d_in[1];
  const float* b1  = (const float*)d_in[2];
  const float* W2  = (const float*)d_in[3];
  const float* b2  = (const float*)d_in[4];
  const float* W3  = (const float*)d_in[5];
  const float* b3  = (const float*)d_in[6];
  const float* W4  = (const float*)d_in[7];
  const float* b4  = (const float*)d_in[8];

  float*  y    = (float*)d_ws;                                   // 128*512 f32
  float4* coef = (float4*)((char*)d_ws + 128 * BATCH * sizeof(float)); // 512*127 f4

  size_t mlp_lds = (size_t)(4 * NP + 2 * NP * NP + NP * N4 + 3 * NP + N4 +
                            8 * 16 * NP) * sizeof(float);        // ~214 KB
  mlp_wmma_kernel<<<8, 128, mlp_lds, stream>>>(phi, W1, b1, W2, b2, W3, b3,
                                               W4, b4, y);

  size_t sp_lds = (size_t)(126 * 128 + 127) * sizeof(float);     // ~65 KB
  spline_kernel<<<4, 128, sp_lds, stream>>>(y, coef);

  eval_kernel<<<256, 256, 0, stream>>>(coef, (float*)d_out);
}